// StereoLoss_21990232555882
// MI455X (gfx1250) — compile-verified
//
#include <hip/hip_runtime.h>
#include <math.h>

// ---------------------------------------------------------------------------
// StereoLoss on MI455X (gfx1250, wave32)
//
// Memory-bound streaming reduction: 248 B + ~450 f32 FLOPs per element
// (~130 MB total -> ~5.6 us floor at 23.3 TB/s). One element per thread,
// coalesced b128 loads, scalar f32 VALU math, then a deterministic
// hierarchical reduction:
//   wave:  V_WMMA_F32_16X16X4_F32 with B = all-ones (layout-invariant)
//          => C[m,n] = lane(m) + lane(m+16); 8 VGPR adds + shfl_xor(16)
//   block: LDS across the 8 waves
//   grid:  per-block partials in d_ws, fixed-order tree in a 2nd kernel
// No float atomics -> bitwise deterministic across graph replays.
// ---------------------------------------------------------------------------

#define BLOCK  256
#define NWAVES (BLOCK / 32)
#define QEPS   1e-8f

typedef __attribute__((ext_vector_type(2))) float v2f;
typedef __attribute__((ext_vector_type(8))) float v8f;

__device__ __forceinline__ void q2r(float w, float x, float y, float z, float* R)
{
    float n = rsqrtf(w*w + x*x + y*y + z*z);
    w *= n; x *= n; y *= n; z *= n;
    R[0] = 1.f - 2.f*(y*y + z*z); R[1] = 2.f*(x*y - z*w);       R[2] = 2.f*(x*z + y*w);
    R[3] = 2.f*(x*y + z*w);       R[4] = 1.f - 2.f*(x*x + z*z); R[5] = 2.f*(y*z - x*w);
    R[6] = 2.f*(x*z - y*w);       R[7] = 2.f*(y*z + x*w);       R[8] = 1.f - 2.f*(x*x + y*y);
}

// Shepperd rotation->quaternion, matching the reference's branch order and
// EPS clamp. Normalization is dropped: _quat_distance is scale-invariant.
__device__ __forceinline__ float4 mat2q(const float* R)
{
    const float r00=R[0], r01=R[1], r02=R[2];
    const float r10=R[3], r11=R[4], r12=R[5];
    const float r20=R[6], r21=R[7], r22=R[8];
    const float tr = r00 + r11 + r22;
    float4 q;
    if (tr > 0.f) {
        float S = 2.f*sqrtf(fmaxf(tr + 1.f, QEPS)), inv = 1.f/S;
        q = make_float4(0.25f*S, (r21-r12)*inv, (r02-r20)*inv, (r10-r01)*inv);
    } else if (r00 > r11 && r00 > r22) {
        float S = 2.f*sqrtf(fmaxf(1.f + r00 - r11 - r22, QEPS)), inv = 1.f/S;
        q = make_float4((r21-r12)*inv, 0.25f*S, (r01+r10)*inv, (r02+r20)*inv);
    } else if (r11 > r22) {
        float S = 2.f*sqrtf(fmaxf(1.f + r11 - r00 - r22, QEPS)), inv = 1.f/S;
        q = make_float4((r02-r20)*inv, (r01+r10)*inv, 0.25f*S, (r12+r21)*inv);
    } else {
        float S = 2.f*sqrtf(fmaxf(1.f + r22 - r00 - r11, QEPS)), inv = 1.f/S;
        q = make_float4((r10-r01)*inv, (r02+r20)*inv, (r12+r21)*inv, 0.25f*S);
    }
    return q;
}

__device__ __forceinline__ void stereo_elem(
    const float4* __restrict__ TIL, const float4* __restrict__ TIR,
    const float4* __restrict__ TGT, const float*  __restrict__ terr,
    const float4* __restrict__ rerr, int b, int B,
    float& lt_out, float& lr_out)
{
    // Row-major 4x4; bottom row is [0,0,0,1] by construction -> only rows 0..2.
    const float4 l0 = TIL[4*b+0], l1 = TIL[4*b+1], l2 = TIL[4*b+2];
    const float4 i0 = TIR[4*b+0], i1 = TIR[4*b+1], i2 = TIR[4*b+2];
    const float4 g0 = TGT[4*b+0], g1 = TGT[4*b+1], g2 = TGT[4*b+2];
    const float tl0 = terr[3*b+0], tl1 = terr[3*b+1], tl2 = terr[3*b+2];
    const int   rb  = 3*(B + b);
    const float tq0 = terr[rb+0], tq1 = terr[rb+1], tq2 = terr[rb+2];
    const float4 ql = rerr[b], qr = rerr[B + b];

    float Rl[9], Rr[9];
    q2r(ql.x, ql.y, ql.z, ql.w, Rl);
    q2r(qr.x, qr.y, qr.z, qr.w, Rr);

    const float Mil[9] = { l0.x,l0.y,l0.z, l1.x,l1.y,l1.z, l2.x,l2.y,l2.z };
    const float Mir[9] = { i0.x,i0.y,i0.z, i1.x,i1.y,i1.z, i2.x,i2.y,i2.z };

    // T_cam2 = inv([Rl,tl]) @ T_init_left = [Rl^T Mil, Rl^T (t_il - tl)]
    // T_cam3 = [Rr^T Mir, Rr^T (t_ir - tr)] = [Bm, bt]
    float A[9], Bm[9];
    #pragma unroll
    for (int i = 0; i < 3; ++i)
        #pragma unroll
        for (int j = 0; j < 3; ++j) {
            A[3*i+j]  = Rl[i]*Mil[j] + Rl[3+i]*Mil[3+j] + Rl[6+i]*Mil[6+j];
            Bm[3*i+j] = Rr[i]*Mir[j] + Rr[3+i]*Mir[3+j] + Rr[6+i]*Mir[6+j];
        }
    const float dl0 = l0.w - tl0, dl1 = l1.w - tl1, dl2 = l2.w - tl2;
    const float dr0 = i0.w - tq0, dr1 = i1.w - tq1, dr2 = i2.w - tq2;
    float at[3], bt[3];
    #pragma unroll
    for (int i = 0; i < 3; ++i) {
        at[i] = Rl[i]*dl0 + Rl[3+i]*dl1 + Rl[6+i]*dl2;
        bt[i] = Rr[i]*dr0 + Rr[3+i]*dr1 + Rr[6+i]*dr2;
    }

    // T23_pred = T_cam2 @ inv(T_cam3) = [A Bm^T, at - A (Bm^T bt)]
    float Rp[9], u[3];
    #pragma unroll
    for (int i = 0; i < 3; ++i) {
        u[i] = Bm[i]*bt[0] + Bm[3+i]*bt[1] + Bm[6+i]*bt[2];
        #pragma unroll
        for (int j = 0; j < 3; ++j)
            Rp[3*i+j] = A[3*i]*Bm[3*j] + A[3*i+1]*Bm[3*j+1] + A[3*i+2]*Bm[3*j+2];
    }
    float tp[3];
    #pragma unroll
    for (int i = 0; i < 3; ++i)
        tp[i] = at[i] - (A[3*i]*u[0] + A[3*i+1]*u[1] + A[3*i+2]*u[2]);

    // translation smooth-L1
    const float tg[3] = { g0.w, g1.w, g2.w };
    float lt = 0.f;
    #pragma unroll
    for (int i = 0; i < 3; ++i) {
        float d = tp[i] - tg[i], ad = fabsf(d);
        lt += (ad < 1.f) ? 0.5f*d*d : (ad - 0.5f);
    }

    // quaternion geodesic distance (scale-invariant in both quats)
    const float G[9] = { g0.x,g0.y,g0.z, g1.x,g1.y,g1.z, g2.x,g2.y,g2.z };
    const float4 qp = mat2q(Rp);
    const float4 qg = mat2q(G);
    const float rn = qg.x*qg.x + qg.y*qg.y + qg.z*qg.z + qg.w*qg.w;
    const float inv = 1.f / rn;
    const float bw =  qg.x*inv, bx = -qg.y*inv, by = -qg.z*inv, bz = -qg.w*inv;
    const float aw = qp.x, ax = qp.y, ay = qp.z, az = qp.w;
    const float tw = aw*bw - ax*bx - ay*by - az*bz;
    const float tx = aw*bx + ax*bw + ay*bz - az*by;
    const float ty = aw*by - ax*bz + ay*bw + az*bx;
    const float tz = aw*bz + ax*by - ay*bx + az*bw;

    lt_out = lt;
    lr_out = 2.f*atan2f(sqrtf(tx*tx + ty*ty + tz*tz), fabsf(tw));
}

// Main kernel: grid covers exactly gridDim.x*BLOCK elements -> EXEC all-1s,
// so the WMMA-based wave reduction is legal.
__global__ void __launch_bounds__(BLOCK)
stereo_loss_main(const float4* __restrict__ TIL, const float4* __restrict__ TIR,
                 const float4* __restrict__ TGT, const float*  __restrict__ terr,
                 const float4* __restrict__ rerr, int B,
                 float* __restrict__ partials)
{
    const int b = blockIdx.x * BLOCK + threadIdx.x;
    float lt, lr;
    stereo_elem(TIL, TIR, TGT, terr, rerr, b, B, lt, lr);

    // --- wave32 reduction via V_WMMA_F32_16X16X4_F32 ---
    // A 16x4 f32 layout (ISA 7.12.2): lane<16 row M=lane, a[0]->K0, a[1]->K1;
    // lane>=16 row M=lane-16, a[0]->K2, a[1]->K3.  B := all-ones is layout-
    // invariant, so C[m,n] = val(lane m) + val(lane m+16).
    v2f ones; ones[0] = 1.f; ones[1] = 1.f;
    v2f av;   av[1] = 0.f;
    v8f c0 = {}, c1 = {};
    av[0] = lt;
    c0 = __builtin_amdgcn_wmma_f32_16x16x4_f32(false, av, false, ones,
                                               (short)0, c0, false, false);
    av[0] = lr;
    c1 = __builtin_amdgcn_wmma_f32_16x16x4_f32(false, av, false, ones,
                                               (short)0, c1, false, false);
    // Lane L holds C rows m=0..7 (L<16) or m=8..15 (L>=16) in its 8 VGPRs.
    float st = ((c0[0]+c0[1]) + (c0[2]+c0[3])) + ((c0[4]+c0[5]) + (c0[6]+c0[7]));
    float sr = ((c1[0]+c1[1]) + (c1[2]+c1[3])) + ((c1[4]+c1[5]) + (c1[6]+c1[7]));
    st += __shfl_xor(st, 16, 32);   // combine the two half-wave row groups
    sr += __shfl_xor(sr, 16, 32);

    __shared__ float smem[2 * NWAVES];
    const int lane = threadIdx.x & 31, wv = threadIdx.x >> 5;
    if (lane == 0) { smem[2*wv] = st; smem[2*wv+1] = sr; }
    __syncthreads();
    if (threadIdx.x == 0) {
        float bt = 0.f, br = 0.f;
        #pragma unroll
        for (int w = 0; w < NWAVES; ++w) { bt += smem[2*w]; br += smem[2*w+1]; }
        partials[2*blockIdx.x]     = bt;
        partials[2*blockIdx.x + 1] = br;
    }
}

// Tail kernel (masked lanes possible -> shuffle reduction, no WMMA).
__global__ void __launch_bounds__(BLOCK)
stereo_loss_tail(const float4* __restrict__ TIL, const float4* __restrict__ TIR,
                 const float4* __restrict__ TGT, const float*  __restrict__ terr,
                 const float4* __restrict__ rerr, int B, int start,
                 float* __restrict__ partial_slot)
{
    const int b = start + (int)threadIdx.x;
    float lt = 0.f, lr = 0.f;
    if (b < B) stereo_elem(TIL, TIR, TGT, terr, rerr, b, B, lt, lr);
    #pragma unroll
    for (int m = 16; m > 0; m >>= 1) {
        lt += __shfl_xor(lt, m, 32);
        lr += __shfl_xor(lr, m, 32);
    }
    __shared__ float smem[2 * NWAVES];
    const int lane = threadIdx.x & 31, wv = threadIdx.x >> 5;
    if (lane == 0) { smem[2*wv] = lt; smem[2*wv+1] = lr; }
    __syncthreads();
    if (threadIdx.x == 0) {
        float bt = 0.f, br = 0.f;
        #pragma unroll
        for (int w = 0; w < NWAVES; ++w) { bt += smem[2*w]; br += smem[2*w+1]; }
        partial_slot[0] = bt;
        partial_slot[1] = br;
    }
}

// Fixed-order final reduction of P {transl, rot} partial pairs -> 3 outputs.
__global__ void __launch_bounds__(BLOCK)
stereo_finalize(const float* __restrict__ partials, int P, float invB,
                float* __restrict__ out)
{
    __shared__ float s_t[BLOCK], s_r[BLOCK];
    const int tid = threadIdx.x;
    float lt = 0.f, lr = 0.f;
    for (int i = tid; i < P; i += BLOCK) {
        lt += partials[2*i];
        lr += partials[2*i + 1];
    }
    s_t[tid] = lt; s_r[tid] = lr;
    __syncthreads();
    for (int s = BLOCK >> 1; s > 0; s >>= 1) {
        if (tid < s) { s_t[tid] += s_t[tid + s]; s_r[tid] += s_r[tid + s]; }
        __syncthreads();
    }
    if (tid == 0) {
        const float lt_mean = s_t[0] * invB;
        const float lr_mean = s_r[0] * invB;
        out[0] = lt_mean + lr_mean;  // rescale_trans = rescale_rot = 1.0
        out[1] = lt_mean;
        out[2] = lr_mean;
    }
}

extern "C" void kernel_launch(void* const* d_in, const int* in_sizes, int n_in,
                              void* d_out, int out_size, void* d_ws, size_t ws_size,
                              hipStream_t stream)
{
    const float4* TIL  = (const float4*)d_in[0];   // (B,4,4)
    const float4* TIR  = (const float4*)d_in[1];   // (B,4,4)
    const float4* TGT  = (const float4*)d_in[2];   // (B,4,4)
    const float*  terr = (const float*)d_in[3];    // (2B,3)
    const float4* rerr = (const float4*)d_in[4];   // (2B,4)
    float* out = (float*)d_out;

    const int B = in_sizes[0] / 16;
    float* partials = (float*)d_ws;

    const int gridMain = B / BLOCK;
    const int rem      = B - gridMain * BLOCK;
    const int P        = gridMain + (rem ? 1 : 0);

    if (gridMain > 0)
        stereo_loss_main<<<gridMain, BLOCK, 0, stream>>>(
            TIL, TIR, TGT, terr, rerr, B, partials);
    if (rem > 0)
        stereo_loss_tail<<<1, BLOCK, 0, stream>>>(
            TIL, TIR, TGT, terr, rerr, B, gridMain * BLOCK,
            partials + 2 * gridMain);

    stereo_finalize<<<1, BLOCK, 0, stream>>>(partials, P, 1.0f / (float)B, out);
}